// Repro_56057913147434
// MI455X (gfx1250) — compile-verified
//
#include <hip/hip_runtime.h>
#include <cstdint>
#include <cstddef>

#define HH 345
#define WW 456
#define BLK_X 32
#define BLK_Y 8
#define LDS_FLOATS 12288   // 48 KB: 3-channel packed bounding-box planes

typedef unsigned int v4u __attribute__((ext_vector_type(4)));
typedef int          v8i __attribute__((ext_vector_type(8)));
typedef int          v4i __attribute__((ext_vector_type(4)));

__device__ __forceinline__ void cubic_w(float t, float w[4]) {
    const float A = -0.75f;
    float t1 = t + 1.0f, t2 = 1.0f - t, t3 = 2.0f - t;
    w[0] = ((A * t1 - 5.0f * A) * t1 + 8.0f * A) * t1 - 4.0f * A;
    w[1] = ((A + 2.0f) * t - (A + 3.0f)) * t * t + 1.0f;
    w[2] = ((A + 2.0f) * t2 - (A + 3.0f)) * t2 * t2 + 1.0f;
    w[3] = ((A * t3 - 5.0f * A) * t3 + 8.0f * A) * t3 - 4.0f * A;
}

__global__ __launch_bounds__(256) void bicubic_affine_tdm(
    const float* __restrict__ img,     // (B,3,H,W)
    const float* __restrict__ theta,   // (B,2,3)
    float* __restrict__ out)           // (B,3,H,W)
{
    __shared__ float tile[LDS_FLOATS];

    const int b   = blockIdx.z;
    const int tx0 = blockIdx.x * BLK_X;
    const int ty0 = blockIdx.y * BLK_Y;

    const float t00 = theta[b * 6 + 0], t01 = theta[b * 6 + 1], t02 = theta[b * 6 + 2];
    const float t10 = theta[b * 6 + 3], t11 = theta[b * 6 + 4], t12 = theta[b * 6 + 5];

    auto gxf  = [] (int x) { return (float)x * (2.0f / WW) + (1.0f / WW - 1.0f); };
    auto gyf  = [] (int y) { return (float)y * (2.0f / HH) + (1.0f / HH - 1.0f); };
    auto mapx = [&](float gx, float gy) {
        return (t00 * gx + t01 * gy + t02) * (WW * 0.5f) + (WW * 0.5f - 0.5f);
    };
    auto mapy = [&](float gx, float gy) {
        return (t10 * gx + t11 * gy + t12) * (HH * 0.5f) + (HH * 0.5f - 0.5f);
    };

    // ---- Tile input bounding box from the 4 output-tile corners (uniform) ----
    const float cgx0 = gxf(tx0), cgx1 = gxf(tx0 + BLK_X - 1);
    const float cgy0 = gyf(ty0), cgy1 = gyf(ty0 + BLK_Y - 1);
    float ix0 = mapx(cgx0, cgy0), ix1 = mapx(cgx1, cgy0);
    float ix2 = mapx(cgx0, cgy1), ix3 = mapx(cgx1, cgy1);
    float iy0 = mapy(cgx0, cgy0), iy1 = mapy(cgx1, cgy0);
    float iy2 = mapy(cgx0, cgy1), iy3 = mapy(cgx1, cgy1);
    float ixmin = fminf(fminf(ix0, ix1), fminf(ix2, ix3));
    float ixmax = fmaxf(fmaxf(ix0, ix1), fmaxf(ix2, ix3));
    float iymin = fminf(fminf(iy0, iy1), fminf(iy2, iy3));
    float iymax = fmaxf(fmaxf(iy0, iy1), fmaxf(iy2, iy3));
    auto safefloor = [](float v) { return (int)floorf(fmaxf(fminf(v, 1.0e6f), -1.0e6f)); };
    // tap footprint [-1,+2] plus 1 px safety margin each side
    int xtmin = safefloor(ixmin) - 2, xtmax = safefloor(ixmax) + 3;
    int ytmin = safefloor(iymin) - 2, ytmax = safefloor(iymax) + 3;
    const int bx0 = min(max(xtmin, 0), WW - 1);
    const int bx1 = min(max(xtmax, 0), WW - 1);
    const int by0 = min(max(ytmin, 0), HH - 1);
    const int by1 = min(max(ytmax, 0), HH - 1);
    const int bw = bx1 - bx0 + 1;
    const int bh = by1 - by0 + 1;
    const int plane = bw * bh;
    const bool useLds = (3 * plane <= LDS_FLOATS);   // uniform across the block

    // ---- Wave 0: TDM DMA the 3 channel boxes into LDS (2D descriptors) ----
    const int flat = threadIdx.y * BLK_X + threadIdx.x;
    if (useLds && flat < 32) {
        const uint64_t imgBase = (uint64_t)(uintptr_t)img;
        const uint32_t ldsBase = (uint32_t)(uintptr_t)(&tile[0]);
        const uint32_t td0 = (uint32_t)(WW - bx0);      // remaining extent from tile start
        const uint32_t td1 = (uint32_t)(HH - by0);
        const uint32_t tile0 = (uint32_t)bw;
        const uint32_t tile1 = (uint32_t)bh;
        const uint64_t stride0 = (uint64_t)WW;          // elements between rows
#pragma unroll
        for (int c = 0; c < 3; ++c) {
            const uint64_t ga = imgBase +
                4ull * (((uint64_t)(b * 3 + c) * HH + (uint64_t)by0) * WW + (uint64_t)bx0);
            const uint32_t la = ldsBase + 4u * (uint32_t)(c * plane);
            v4u g0;
            g0.x = 1u;                                   // count=1, user descriptor
            g0.y = la;                                   // lds_addr (bytes)
            g0.z = (uint32_t)ga;                         // global_addr[31:0]
            g0.w = (uint32_t)((ga >> 32) & 0x1FFFFFFu) | 0x80000000u; // addr[56:32] | type=2
            v8i g1;
            g1[0] = (int)(2u << 16);                     // data_size=4B, mask=0, no flags
            g1[1] = (int)((td0 & 0xFFFFu) << 16);        // tensor_dim0[15:0]
            g1[2] = (int)((td0 >> 16) | ((td1 & 0xFFFFu) << 16));
            g1[3] = (int)((td1 >> 16) | (tile0 << 16));  // tile_dim0
            g1[4] = (int)(tile1);                        // tile_dim1, tile_dim2=0 (2D)
            g1[5] = (int)(uint32_t)(stride0 & 0xFFFFFFFFull);
            g1[6] = 0;                                   // stride0 hi, stride1 lo (unused, 2D)
            g1[7] = 0;
            v4i gz4  = {0, 0, 0, 0};
            v8i gz8  = {0, 0, 0, 0, 0, 0, 0, 0};
            __builtin_amdgcn_tensor_load_to_lds(g0, g1, gz4, gz4, gz8, 0);
        }
        __builtin_amdgcn_s_wait_tensorcnt(0);
    }
    __syncthreads();

    // ---- Per-pixel bicubic ----
    const int x = tx0 + (int)threadIdx.x;
    const int y = ty0 + (int)threadIdx.y;
    if (x < WW && y < HH) {
        const float ix = mapx(gxf(x), gyf(y));
        const float iy = mapy(gxf(x), gyf(y));
        const float fx = floorf(ix), fy = floorf(iy);
        float wx[4], wy[4];
        cubic_w(ix - fx, wx);
        cubic_w(iy - fy, wy);
        const int xb = safefloor(fx) - 1;
        const int yb = safefloor(fy) - 1;

        float a0 = 0.0f, a1 = 0.0f, a2 = 0.0f;
        if (useLds) {
#pragma unroll
            for (int j = 0; j < 4; ++j) {
                const int yj = yb + j;
                const bool my = (yj >= 0) && (yj < HH);
                const int yc = min(max(yj, 0), HH - 1);
                const int ly = min(max(yc - by0, 0), bh - 1);
                const float wyv = wy[j];
#pragma unroll
                for (int i = 0; i < 4; ++i) {
                    const int xi = xb + i;
                    const bool m = my && (xi >= 0) && (xi < WW);
                    const int xc = min(max(xi, 0), WW - 1);
                    const int lx = min(max(xc - bx0, 0), bw - 1);
                    const float w = m ? (wyv * wx[i]) : 0.0f;
                    const int o = ly * bw + lx;
                    a0 += w * tile[o];
                    a1 += w * tile[plane + o];
                    a2 += w * tile[2 * plane + o];
                }
            }
        } else {
            const float* __restrict__ p = img + (size_t)b * 3 * HH * WW;
#pragma unroll
            for (int j = 0; j < 4; ++j) {
                const int yj = yb + j;
                const bool my = (yj >= 0) && (yj < HH);
                const int yc = min(max(yj, 0), HH - 1);
                const float wyv = wy[j];
#pragma unroll
                for (int i = 0; i < 4; ++i) {
                    const int xi = xb + i;
                    const bool m = my && (xi >= 0) && (xi < WW);
                    const int xc = min(max(xi, 0), WW - 1);
                    const float w = m ? (wyv * wx[i]) : 0.0f;
                    const size_t o = (size_t)yc * WW + xc;
                    a0 += w * p[o];
                    a1 += w * p[(size_t)HH * WW + o];
                    a2 += w * p[2 * (size_t)HH * WW + o];
                }
            }
        }
        const size_t ob = ((size_t)(b * 3) * HH + (size_t)y) * WW + (size_t)x;
        out[ob]                        = a0;
        out[ob + (size_t)HH * WW]      = a1;
        out[ob + 2 * (size_t)HH * WW]  = a2;
    }
}

extern "C" void kernel_launch(void* const* d_in, const int* in_sizes, int n_in,
                              void* d_out, int out_size, void* d_ws, size_t ws_size,
                              hipStream_t stream) {
    (void)n_in; (void)d_ws; (void)ws_size; (void)out_size;
    const float* img   = (const float*)d_in[1];
    const float* theta = (const float*)d_in[2];
    float* out = (float*)d_out;
    const int B = in_sizes[1] / (3 * HH * WW);
    dim3 grid((WW + BLK_X - 1) / BLK_X, (HH + BLK_Y - 1) / BLK_Y, B);
    dim3 block(BLK_X, BLK_Y, 1);
    bicubic_affine_tdm<<<grid, block, 0, stream>>>(img, theta, out);
}